// PotentialEncoder_68959994904856
// MI455X (gfx1250) — compile-verified
//
#include <hip/hip_runtime.h>
#include <cstdint>

// Multi-resolution hash-grid encode (instant-NGP style), 3 tables.
// Static config from reference: L=16, C=2, D=3, H=16, S=1, MAX_PARAMS=2^19.
//   scale_l = 16*2^l - 1          (exact in f32)
//   R_l     = 16*2^l + 1
//   sizes: l0=4920, l1=35944, l2=274632, l>=3: 524288 (=2^19)
//   offsets: 0, 4920, 40864, then 315496 + (l-3)*524288
//   levels 0..2 tiled (index provably < size, modulo is identity),
//   levels >=3 hashed with primes (1, 2654435761, 805459861), mask 0x7FFFF.
//
// Layout: tid = b*16 + l. Each thread writes out[b*96 + l*6 .. +5], so a
// wave32 writes one contiguous 768B span -> coalesced HBM stores.
// Block = 256 threads = 16 points; the 48 input floats (192B) are staged
// into LDS with gfx1250 async-to-LDS b128 copies + s_wait_asynccnt.
//
// The 3 embedding tables (171 MB) fit in the 192 MB L2; output stores
// (201 MB stream) use non-temporal hints so they don't evict the tables.

typedef float v2f __attribute__((ext_vector_type(2)));

__global__ __launch_bounds__(256)
void grid_encode3_kernel(const float*  __restrict__ inp,
                         const float2* __restrict__ embx,
                         const float2* __restrict__ emby,
                         const float2* __restrict__ embz,
                         float*        __restrict__ out)
{
    __shared__ __align__(16) float s_in[48];   // 16 points * 3 coords

    // ---- async stage of this block's coords: 12 lanes x b128 = 192B ----
    if (threadIdx.x < 12u) {
        const float* gsrc = inp + (size_t)blockIdx.x * 48u + (size_t)threadIdx.x * 4u;
        unsigned lds_addr = (unsigned)(uintptr_t)(const void*)&s_in[threadIdx.x * 4u];
        asm volatile("global_load_async_to_lds_b128 %0, %1, off"
                     :: "v"(lds_addr), "v"(gsrc) : "memory");
        asm volatile("s_wait_asynccnt 0" ::: "memory");
    }
    __syncthreads();

    const unsigned tid = blockIdx.x * 256u + threadIdx.x;
    const unsigned pb  = threadIdx.x >> 4;      // point within block (0..15)
    const unsigned l   = threadIdx.x & 15u;     // level (0..15)

    // normalize: x = (v + 1) * 0.5   (bound == 1)
    const float vx = s_in[pb * 3 + 0];
    const float vy = s_in[pb * 3 + 1];
    const float vz = s_in[pb * 3 + 2];
    const float xn = __fmul_rn(__fadd_rn(vx, 1.0f), 0.5f);
    const float yn = __fmul_rn(__fadd_rn(vy, 1.0f), 0.5f);
    const float zn = __fmul_rn(__fadd_rn(vz, 1.0f), 0.5f);

    // pos = x*scale + 0.5 ; pg = floor(pos) ; frac = pos - pg
    const float scale = (float)((16u << l) - 1u);      // exact (< 2^24)
    const float px = __fadd_rn(__fmul_rn(xn, scale), 0.5f);
    const float py = __fadd_rn(__fmul_rn(yn, scale), 0.5f);
    const float pz = __fadd_rn(__fmul_rn(zn, scale), 0.5f);
    const float gx = floorf(px), gy = floorf(py), gz = floorf(pz);
    const float fx = __fsub_rn(px, gx);
    const float fy = __fsub_rn(py, gy);
    const float fz = __fsub_rn(pz, gz);
    const unsigned pgx = (unsigned)gx, pgy = (unsigned)gy, pgz = (unsigned)gz;

    // per-level constants, all in ALU (no LUT loads)
    const bool     useHash = (l >= 3u);
    const unsigned offset  = useHash ? (315496u + (l - 3u) * 524288u)
                                     : (l == 0u ? 0u : (l == 1u ? 4920u : 40864u));
    const unsigned R  = (16u << l) + 1u;   // only meaningful for l<3
    const unsigned R2 = R * R;

    const float wx0 = __fsub_rn(1.0f, fx);
    const float wy0 = __fsub_rn(1.0f, fy);
    const float wz0 = __fsub_rn(1.0f, fz);

    unsigned idx[8];
    float    w[8];
#pragma unroll
    for (int corner = 0; corner < 8; ++corner) {
        const unsigned bx =  corner       & 1;
        const unsigned by = (corner >> 1) & 1;
        const unsigned bz = (corner >> 2) & 1;
        const unsigned c0 = pgx + bx;
        const unsigned c1 = pgy + by;
        const unsigned c2 = pgz + bz;
        const float w0 = bx ? fx : wx0;
        const float w1 = by ? fy : wy0;
        const float w2 = bz ? fz : wz0;
        w[corner] = __fmul_rn(__fmul_rn(w0, w1), w2);
        const unsigned tiled  = c0 + c1 * R + c2 * R2;
        const unsigned hashed = c0 ^ (c1 * 2654435761u) ^ (c2 * 805459861u);
        idx[corner] = (useHash ? (hashed & 0x7FFFFu) : tiled) + offset;
    }

    // issue all 24 b64 gathers before accumulation (hide L2 latency)
    float2 ex[8], ey[8], ez[8];
#pragma unroll
    for (int corner = 0; corner < 8; ++corner) {
        ex[corner] = embx[idx[corner]];
        ey[corner] = emby[idx[corner]];
        ez[corner] = embz[idx[corner]];
    }

    float ax0 = 0.f, ax1 = 0.f, ay0 = 0.f, ay1 = 0.f, az0 = 0.f, az1 = 0.f;
#pragma unroll
    for (int corner = 0; corner < 8; ++corner) {
        const float wc = w[corner];
        ax0 = __fadd_rn(ax0, __fmul_rn(wc, ex[corner].x));
        ax1 = __fadd_rn(ax1, __fmul_rn(wc, ex[corner].y));
        ay0 = __fadd_rn(ay0, __fmul_rn(wc, ey[corner].x));
        ay1 = __fadd_rn(ay1, __fmul_rn(wc, ey[corner].y));
        az0 = __fadd_rn(az0, __fmul_rn(wc, ez[corner].x));
        az1 = __fadd_rn(az1, __fmul_rn(wc, ez[corner].y));
    }

    // out[b, l*2+c, t] flat = b*96 + l*6 + c*3 + t = tid*6 + c*3 + t
    // order: [x.c0, y.c0, z.c0, x.c1, y.c1, z.c1]; tid*6*4 bytes is 8B-aligned.
    // Non-temporal stores: keep the 201MB output stream from evicting the
    // L2-resident embedding tables (CDNA5 store TH=NT).
    v2f* op = (v2f*)(out + (size_t)tid * 6u);
    v2f o0; o0.x = ax0; o0.y = ay0;
    v2f o1; o1.x = az0; o1.y = ax1;
    v2f o2; o2.x = ay1; o2.y = az1;
    __builtin_nontemporal_store(o0, op + 0);
    __builtin_nontemporal_store(o1, op + 1);
    __builtin_nontemporal_store(o2, op + 2);
}

extern "C" void kernel_launch(void* const* d_in, const int* in_sizes, int n_in,
                              void* d_out, int out_size, void* d_ws, size_t ws_size,
                              hipStream_t stream) {
    const float*  inp  = (const float*)d_in[0];
    const float2* embx = (const float2*)d_in[1];
    const float2* emby = (const float2*)d_in[2];
    const float2* embz = (const float2*)d_in[3];
    float* out = (float*)d_out;

    const int B = in_sizes[0] / 3;         // 524288
    const int blocks = B / 16;             // 16 points per 256-thread block
    grid_encode3_kernel<<<blocks, 256, 0, stream>>>(inp, embx, emby, embz, out);
}